// C2FCritic_4080218931211
// MI455X (gfx1250) — compile-verified
//
#include <hip/hip_runtime.h>
#include <stdint.h>

// C2F critic: C51 distributional projection (per-batch WMMA GEMM) + action decode.
//
// next_q_probs [4096, 3, 6, 9, 51] f32; projection weights shared across the 162
// inner rows of each batch -> M_b(162x51) = P_b(162x51) x W_b(51x51).
// Padded WMMA problem per batch: 176 x 64 x 52 using v_wmma_f32_16x16x4_f32.
// Staging: global -> LDS via GLOBAL_LOAD_ASYNC_TO_LDS_B64 (ASYNCcnt).

#define LEVELS   3
#define BINS     9
#define ATOMS    51
#define ADIM     6
#define ROWS     162                   // LEVELS*ADIM*BINS
#define ROW_ELEMS (ROWS * ATOMS)       // 8262 floats per batch block
#define V_MIN_F  (-10.0f)
#define V_MAX_F  (10.0f)
#define DELTA_Z  (0.4f)                // (V_MAX-V_MIN)/(ATOMS-1)

#define MTILES   11                    // 176 >= 162 rows
#define KSTEPS   13                    // 52  >= 51 K (padded with zeros)
#define WCOLS    64                    // W row stride (dwords)
#define WROWS    52
#define P_PAD_DW (176 * ATOMS + 4)     // 8980 dwords: rows 162..175 + K overrun slack, zeroed

typedef __attribute__((ext_vector_type(2))) float v2f;
typedef __attribute__((ext_vector_type(8))) float v8f;

__global__ __launch_bounds__(128)
void project_wmma_kernel(const float* __restrict__ p,
                         const float* __restrict__ reward,
                         const float* __restrict__ discount,
                         const float* __restrict__ support,
                         float* __restrict__ out)
{
    __shared__ __align__(16) float lds_p[P_PAD_DW];      // 35,920 B staged probs (+zero pad)
    __shared__ __align__(16) float lds_w[WROWS * WCOLS]; // 13,312 B dense projection matrix

    const int b   = blockIdx.x;
    const int tid = threadIdx.x;

    // ---- Stage this batch's 162x51 block: async global -> LDS (no VGPR roundtrip) ----
    {
        const uint64_t gbase   = (uint64_t)(uintptr_t)(p + (size_t)b * ROW_ELEMS);
        const uint32_t ldsbase = (uint32_t)(uintptr_t)&lds_p[0];  // low 32 bits = LDS offset
        for (int c = tid; c < ROW_ELEMS / 2; c += 128) {          // 4131 x 8-byte chunks
            const uint32_t voff  = (uint32_t)c * 8u;
            const uint32_t laddr = ldsbase + voff;
            asm volatile("global_load_async_to_lds_b64 %0, %1, %2"
                         :: "v"(laddr), "v"(voff), "s"(gbase) : "memory");
        }
        // Zero padding region (rows 162..175 + K-overrun slack): exact 0 contributions.
        for (int i = ROW_ELEMS + tid; i < P_PAD_DW; i += 128)
            lds_p[i] = 0.0f;
        // Zero W (row 51 and cols 51..63 stay zero).
        for (int i = tid; i < WROWS * WCOLS; i += 128)
            lds_w[i] = 0.0f;
        asm volatile("s_wait_asynccnt 0x0" ::: "memory");
    }
    __syncthreads();

    // ---- Build W_b: one thread per source atom j ----
    if (tid < ATOMS) {
        const float rew  = reward[b];
        const float disc = discount[b];
        float Tz = fminf(fmaxf(rew + disc * support[tid], V_MIN_F), V_MAX_F);
        float bb = (Tz - V_MIN_F) / DELTA_Z;
        int lo = (int)floorf(bb);
        int up = (int)ceilf(bb);
        // sequential fixups exactly as in the reference (second test sees updated lo)
        if (up > 0 && lo == up) lo -= 1;
        if (lo < ATOMS - 1 && lo == up) up += 1;
        lds_w[tid * WCOLS + lo] += (float)up - bb;
        lds_w[tid * WCOLS + up] += bb - (float)lo;
    }
    __syncthreads();

    // ---- WMMA projection: wave w owns N-tile w; loop M-tiles; chain K in C ----
    const int lane = tid & 31;
    const int wave = tid >> 5;
    const int lh   = lane & 15;
    const int hi   = lane >> 4;          // 0: lanes 0-15, 1: lanes 16-31
    const int n0   = wave * 16;

    // B fragments: VGPR0 = K-row (k0 + 2*hi), VGPR1 = K-row (k0 + 2*hi + 1),
    // column = n0 + (lane&15). Hoisted: W tile reused across all 11 M-tiles.
    v2f bfrag[KSTEPS];
#pragma unroll
    for (int s = 0; s < KSTEPS; ++s) {
        const int r = 4 * s + 2 * hi;
        v2f bf;
        bf.x = lds_w[r * WCOLS + n0 + lh];
        bf.y = lds_w[(r + 1) * WCOLS + n0 + lh];
        bfrag[s] = bf;
    }

    float* __restrict__ outb = out + (size_t)b * ROW_ELEMS;
    const int  col    = n0 + lh;
    const bool col_ok = (col < ATOMS);   // loop-invariant lane predicate

#pragma unroll 1
    for (int mt = 0; mt < MTILES; ++mt) {
        // A fragment: lane holds row M = mt*16 + (lane&15), K = k0 + 2*hi + {0,1}
        const int abase = (mt * 16 + lh) * ATOMS + 2 * hi;
        v8f acc = {0.f, 0.f, 0.f, 0.f, 0.f, 0.f, 0.f, 0.f};
#pragma unroll
        for (int s = 0; s < KSTEPS; ++s) {
            v2f a;
            a.x = lds_p[abase + 4 * s];
            a.y = lds_p[abase + 4 * s + 1];
            // D = A(16x4) x B(4x16) + C  (neg_a, A, neg_b, B, c_mod, C, reuse_a, reuse_b)
            acc = __builtin_amdgcn_wmma_f32_16x16x4_f32(
                false, a, false, bfrag[s], (short)0, acc, false, false);
        }
        // D layout: VGPR v -> row (mt*16 + v + 8*hi), col n0+(lane&15).
        // Rows are always valid for mt<10; for mt==10 only rows 160/161 (hi==0, v<2).
        if (col_ok) {
            if (mt < MTILES - 1) {
                const int rbase = mt * 16 + hi * 8;
#pragma unroll
                for (int v = 0; v < 8; ++v)
                    outb[(rbase + v) * ATOMS + col] = acc[v];
            } else if (hi == 0) {
                outb[160 * ATOMS + col] = acc[0];
                outb[161 * ATOMS + col] = acc[1];
            }
        }
    }
}

// ---- Action encode+decode (tiny, elementwise, bandwidth-irrelevant) ----
__global__ void decode_kernel(const float* __restrict__ action,
                              const float* __restrict__ low0,
                              const float* __restrict__ high0,
                              float* __restrict__ out, int n)
{
    const int i = blockIdx.x * blockDim.x + threadIdx.x;
    if (i >= n) return;
    const int d = i % ADIM;
    const float a = action[i];
    float low = low0[d];
    float high = high0[d];
#pragma unroll
    for (int l = 0; l < LEVELS; ++l) {
        const float slice = (high - low) / (float)BINS;
        float idx = floorf((a - low) / slice);
        idx = fminf(fmaxf(idx, 0.0f), (float)(BINS - 1));
        const float na = fminf(fmaxf(low + slice * idx, -1.0f), 1.0f);
        low  = fmaxf(-1.0f, na);
        high = fminf(1.0f, na + slice);
    }
    out[i] = 0.5f * (high + low);
}

extern "C" void kernel_launch(void* const* d_in, const int* in_sizes, int n_in,
                              void* d_out, int out_size, void* d_ws, size_t ws_size,
                              hipStream_t stream)
{
    const float* p        = (const float*)d_in[0];   // next_q_probs [B,3,6,9,51]
    const float* reward   = (const float*)d_in[1];   // [B,1]
    const float* discount = (const float*)d_in[2];   // [B,1]
    const float* action   = (const float*)d_in[3];   // [B,6]
    const float* support  = (const float*)d_in[4];   // [51]
    const float* low0     = (const float*)d_in[5];   // [6]
    const float* high0    = (const float*)d_in[6];   // [6]

    const int B = in_sizes[1];                       // 4096

    float* proj = (float*)d_out;                     // [B*162*51]
    float* dec  = proj + (size_t)B * ROW_ELEMS;      // [B*6]

    project_wmma_kernel<<<B, 128, 0, stream>>>(p, reward, discount, support, proj);

    const int n = B * ADIM;
    decode_kernel<<<(n + 255) / 256, 256, 0, stream>>>(action, low0, high0, dec, n);
}